// QBN_78374563217451
// MI455X (gfx1250) — compile-verified
//
#include <hip/hip_runtime.h>

typedef float v2f __attribute__((ext_vector_type(2)));
typedef float v8f __attribute__((ext_vector_type(8)));

#define F_DIM 256
#define B_DIM 64
#define Q_DIM 4
#define N_DIM 1024
#define BN_INV (1.0f / (float)(B_DIM * N_DIM))
#define EPSV 1e-5f

// ---------------- zero accumulators ----------------
__global__ void qbn_zero(float* ws, int n) {
  int i = blockIdx.x * blockDim.x + threadIdx.x;
  if (i < n) ws[i] = 0.0f;
}

// ---------------- pass 1: sums + covariance via f32 WMMA ----------------
// One wave handles (f = blockIdx.x, b = blockIdx.y*8 + wave_id), reducing over
// n = 0..1023 in chunks of 16 samples per V_WMMA_F32_16X16X4_F32.
// Lane mapping (wave32): q = lane&3, g = (lane&15)>>2, kb = (lane>>4)*2.
// A[m=4g+i, k] = x[q=i, sample 4g+k]  ->  per-lane float2 at sample 4g+kb(+1).
// B uses the identical per-lane data (ISA 32-bit A(16x4)/B(4x16) layouts match),
// so D's diagonal 4x4 blocks each accumulate Sigma x_i x_j (uncentered) over their group.
__global__ __launch_bounds__(256) void qbn_stats(const float* __restrict__ x,
                                                 float* __restrict__ s1,
                                                 float* __restrict__ vacc) {
  const int f    = blockIdx.x;
  const int lane = threadIdx.x & 31;
  const int wave = threadIdx.x >> 5;
  const int b    = blockIdx.y * 8 + wave;

  const int q     = lane & 3;
  const int g     = (lane & 15) >> 2;
  const int kb    = (lane >> 4) << 1;
  const int s_off = 4 * g + kb;           // always even -> 8B-aligned float2

  const float* p =
      x + (((size_t)b * F_DIM + f) * Q_DIM + q) * (size_t)N_DIM + s_off;

  v8f acc = {};
  float s1l = 0.0f;

  for (int n = 0; n < N_DIM; n += 16) {
    v2f a = *(const v2f*)(p + n);
    __builtin_prefetch(p + n + 256, 0, 1);      // global_prefetch_b8
    s1l += a.x + a.y;
    // D = A*B + C   (A == B: symmetric Gram accumulation, K = 4 per issue,
    // 4 diagonal 4x4 blocks -> 16 samples per WMMA)
    acc = __builtin_amdgcn_wmma_f32_16x16x4_f32(
        /*neg_a=*/false, a, /*neg_b=*/false, a,
        /*c_mod=*/(short)0, acc, /*reuse_a=*/false, /*reuse_b=*/false);
  }

  atomicAdd(&s1[f * Q_DIM + q], s1l);

  // Useful lanes hold diagonal blocks: cols 0..7 with rows 0..7 (lanes 0..7)
  // and cols 8..15 with rows 8..15 (lanes 24..31).
  if (lane < 8 || lane >= 24) {
    const int j     = lane & 3;
    const int rbase = (lane >> 2) & 1;    // 0 -> acc[0..3], 1 -> acc[4..7]
#pragma unroll
    for (int i = 0; i < 4; ++i) {
      float v = rbase ? acc[4 + i] : acc[i];
      atomicAdd(&vacc[f * 16 + i * 4 + j], v);
    }
  }
}

// ---------------- pass 2: per-f 4x4 inverse + Cholesky ----------------
__global__ void qbn_solve(const float* __restrict__ s1,
                          const float* __restrict__ vacc,
                          float* __restrict__ mean,
                          float* __restrict__ wmat) {
  int f = blockIdx.x * blockDim.x + threadIdx.x;
  if (f >= F_DIM) return;

  float E[4];
#pragma unroll
  for (int qq = 0; qq < 4; ++qq) E[qq] = s1[f * 4 + qq] * BN_INV;

  float V[4][4];
#pragma unroll
  for (int i = 0; i < 4; ++i)
#pragma unroll
    for (int j = 0; j < 4; ++j) {
      float v = vacc[f * 16 + i * 4 + j] * BN_INV - E[i] * E[j];
      if (i == j) v += EPSV;
      V[i][j] = v;
    }

  // Gauss-Jordan inverse (SPD: no pivoting needed)
  float Inv[4][4];
#pragma unroll
  for (int i = 0; i < 4; ++i)
#pragma unroll
    for (int j = 0; j < 4; ++j) Inv[i][j] = (i == j) ? 1.0f : 0.0f;

#pragma unroll
  for (int k = 0; k < 4; ++k) {
    float piv = 1.0f / V[k][k];
#pragma unroll
    for (int j = 0; j < 4; ++j) { V[k][j] *= piv; Inv[k][j] *= piv; }
#pragma unroll
    for (int r = 0; r < 4; ++r) {
      if (r == k) continue;
      float m = V[r][k];
#pragma unroll
      for (int j = 0; j < 4; ++j) { V[r][j] -= m * V[k][j]; Inv[r][j] -= m * Inv[k][j]; }
    }
  }

  // Cholesky of Inv (lower)
  float W[4][4] = {};
#pragma unroll
  for (int i = 0; i < 4; ++i) {
#pragma unroll
    for (int j = 0; j < 4; ++j) {
      if (j > i) continue;
      float s = Inv[i][j];
#pragma unroll
      for (int k = 0; k < 4; ++k)
        if (k < j) s -= W[i][k] * W[j][k];
      W[i][j] = (i == j) ? sqrtf(s) : s / W[j][j];
    }
  }

#pragma unroll
  for (int qq = 0; qq < 4; ++qq) mean[f * 4 + qq] = E[qq];
#pragma unroll
  for (int i = 0; i < 4; ++i)
#pragma unroll
    for (int j = 0; j < 4; ++j) wmat[f * 16 + i * 4 + j] = W[i][j];
}

// ---------------- pass 3: whiten + affine ----------------
// blockIdx.x = b*1024 + f*4 + nb ; each block covers 256 consecutive n of one
// (b,f) -> f is block-uniform so W/E/gamma/beta loads scalarize.
__global__ __launch_bounds__(256) void qbn_apply(const float* __restrict__ x,
                                                 const float* __restrict__ gamma,
                                                 const float* __restrict__ beta,
                                                 const float* __restrict__ mean,
                                                 const float* __restrict__ wmat,
                                                 float* __restrict__ out) {
  const int f = (blockIdx.x >> 2) & (F_DIM - 1);
  const int b = blockIdx.x >> 10;
  const int n = ((blockIdx.x & 3) << 8) | threadIdx.x;

  const size_t base = ((size_t)b * F_DIM + f) * (size_t)(Q_DIM * N_DIM) + n;

  float Wl[4][4];
#pragma unroll
  for (int i = 0; i < 4; ++i)
#pragma unroll
    for (int j = 0; j < 4; ++j) Wl[i][j] = wmat[f * 16 + i * 4 + j];

  float xc[4];
#pragma unroll
  for (int qq = 0; qq < 4; ++qq)
    xc[qq] = x[base + (size_t)qq * N_DIM] - mean[f * 4 + qq];

#pragma unroll
  for (int i = 0; i < 4; ++i) {
    float s = 0.0f;
#pragma unroll
    for (int j = 0; j < 4; ++j)
      if (j <= i) s = fmaf(Wl[i][j], xc[j], s);
    out[base + (size_t)i * N_DIM] = fmaf(gamma[f * 4 + i], s, beta[f * 4 + i]);
  }
}

// ---------------- launch ----------------
extern "C" void kernel_launch(void* const* d_in, const int* in_sizes, int n_in,
                              void* d_out, int out_size, void* d_ws, size_t ws_size,
                              hipStream_t stream) {
  const float* x     = (const float*)d_in[0];
  const float* gamma = (const float*)d_in[1];
  const float* beta  = (const float*)d_in[2];
  float* out = (float*)d_out;

  float* ws   = (float*)d_ws;
  float* s1   = ws;                         // F*4   = 1024 floats
  float* vacc = s1 + F_DIM * Q_DIM;         // F*16  = 4096 floats
  float* mean = vacc + F_DIM * 16;          // F*4   = 1024 floats
  float* wmat = mean + F_DIM * Q_DIM;       // F*16  = 4096 floats

  const int zn = F_DIM * Q_DIM + F_DIM * 16;  // zero s1 + vacc each call
  qbn_zero<<<(zn + 255) / 256, 256, 0, stream>>>(ws, zn);
  qbn_stats<<<dim3(F_DIM, B_DIM / 8), 256, 0, stream>>>(x, s1, vacc);
  qbn_solve<<<1, 256, 0, stream>>>(s1, vacc, mean, wmat);
  qbn_apply<<<B_DIM * F_DIM * (N_DIM / 256), 256, 0, stream>>>(
      x, gamma, beta, mean, wmat, out);
}